// GCNLinkPredictor_13855564497404
// MI455X (gfx1250) — compile-verified
//
#include <hip/hip_runtime.h>

typedef float v2f __attribute__((ext_vector_type(2)));
typedef float v8f __attribute__((ext_vector_type(8)));

// ---------------------------------------------------------------- degree prep
__global__ void fill_ones_kernel(float* __restrict__ p, int n) {
    int i = blockIdx.x * blockDim.x + threadIdx.x;
    if (i < n) p[i] = 1.0f;   // self-loop contributes 1 to every degree
}

__global__ void deg_count_kernel(const int* __restrict__ dst, int E,
                                 float* __restrict__ deg) {
    int e = blockIdx.x * blockDim.x + threadIdx.x;
    if (e < E) atomicAdd(&deg[dst[e]], 1.0f);
}

__global__ void rsqrt_kernel(float* __restrict__ p, int n) {
    int i = blockIdx.x * blockDim.x + threadIdx.x;
    if (i < n) p[i] = rsqrtf(p[i]);   // deg >= 1 always (self-loop)
}

// ---------------------------------------------------------------- f32 WMMA GEMM
// Out[M,Cout] = act(H[M,K]) @ W[K,Cout].  One wave per 16x16 tile.
// A-frag layout (V_WMMA_F32_16X16X4_F32): lanes 0-15 row M=l hold K=0,1;
// lanes 16-31 hold K=2,3.  B-frag mirrors it (K rows striped over halves).
template <bool RELU_IN>
__global__ void gemm_wmma_f32_kernel(const float* __restrict__ H,
                                     const float* __restrict__ W,
                                     float* __restrict__ Out,
                                     int M, int K, int Cout) {
    int wave = (int)((blockIdx.x * blockDim.x + threadIdx.x) >> 5);
    int lane = threadIdx.x & 31;
    int ntn  = Cout >> 4;
    int tiles = (M >> 4) * ntn;          // M, Cout are multiples of 16 here
    if (wave >= tiles) return;           // wave-uniform: EXEC stays all-ones

    int mt = wave / ntn;
    int nt = wave - mt * ntn;
    int half = lane >> 4;                // 0: K pair {0,1}, 1: K pair {2,3}
    int l    = lane & 15;

    const float* hrow = H + (size_t)((mt << 4) + l) * K;
    int ncol = (nt << 4) + l;

    v8f c = {0.f, 0.f, 0.f, 0.f, 0.f, 0.f, 0.f, 0.f};

    #pragma unroll 4
    for (int k0 = 0; k0 < K; k0 += 4) {
        int ka = k0 + half * 2;
        v2f a = *(const v2f*)(hrow + ka);          // 8B-aligned (K mult of 4)
        if (RELU_IN) { a.x = fmaxf(a.x, 0.f); a.y = fmaxf(a.y, 0.f); }
        v2f b;
        b.x = W[(size_t)ka * Cout + ncol];
        b.y = W[(size_t)(ka + 1) * Cout + ncol];
        c = __builtin_amdgcn_wmma_f32_16x16x4_f32(
                /*neg_a=*/false, a, /*neg_b=*/false, b,
                /*c_mod=*/(short)0, c, /*reuse_a=*/false, /*reuse_b=*/false);
    }

    // C/D layout: VGPR v -> M = v (lanes 0-15) or v+8 (lanes 16-31), N = l.
    #pragma unroll
    for (int v = 0; v < 8; ++v) {
        int row = (mt << 4) + v + half * 8;
        Out[(size_t)row * Cout + ncol] = c[v];
    }
}

// ------------------------------------------------- self-loop + bias initializer
// O[i,c] = T[i,c] * dinv[i]^2 + bias[c]   (float4-vectorized, C compile-time)
template <int C>
__global__ void self_bias_kernel(const float4* __restrict__ T,
                                 const float* __restrict__ dinv,
                                 const float4* __restrict__ bias,
                                 float4* __restrict__ O, int total4) {
    int idx = blockIdx.x * blockDim.x + threadIdx.x;
    if (idx >= total4) return;
    constexpr int C4 = C / 4;
    int i  = idx / C4;               // power-of-two -> shift
    int cc = idx & (C4 - 1);
    float di = dinv[i];
    float s = di * di;
    float4 t = T[idx];
    float4 b = bias[cc];
    float4 o;
    o.x = t.x * s + b.x;  o.y = t.y * s + b.y;
    o.z = t.z * s + b.z;  o.w = t.w * s + b.w;
    O[idx] = o;
}

// ----------------------------------------------------- edge scatter (segment_sum)
// O[dst] += T[src] * dinv[src]*dinv[dst].  float4 chunk per lane;
// C=128 -> 1 edge/wave (32 lanes), C=64 -> 2 edges/wave (16 lanes each).
template <int C>
__global__ void scatter_edges_kernel(const float* __restrict__ T,
                                     const float* __restrict__ dinv,
                                     const int* __restrict__ src,
                                     const int* __restrict__ dst,
                                     float* __restrict__ O, int E) {
    constexpr int NCH = C / 4;       // float4 chunks per edge (32 or 16)
    constexpr int EPW = 32 / NCH;    // edges per wave (1 or 2)
    int gw   = (int)((blockIdx.x * blockDim.x + threadIdx.x) >> 5);
    int lane = threadIdx.x & 31;
    int e = gw * EPW + lane / NCH;
    int l = lane & (NCH - 1);
    if (e >= E) return;
    int s = src[e], d = dst[e];
    float nm = dinv[s] * dinv[d];
    float4 v = ((const float4*)(T + (size_t)s * C))[l];
    float* od = O + (size_t)d * C + l * 4;
    atomicAdd(od + 0, v.x * nm);
    atomicAdd(od + 1, v.y * nm);
    atomicAdd(od + 2, v.z * nm);
    atomicAdd(od + 3, v.w * nm);
}

// ---------------------------------------------------------------- edge decode
// C = 64: half-wave (16 lanes) per edge, one float4 per lane per operand.
__global__ void decode_edges_kernel(const float* __restrict__ Z,
                                    const int* __restrict__ src,
                                    const int* __restrict__ dst,
                                    float* __restrict__ out, int E) {
    int he = (int)((blockIdx.x * blockDim.x + threadIdx.x) >> 4);  // half-wave id
    int l  = threadIdx.x & 15;
    if (he >= E) return;
    int s = src[he], d = dst[he];
    float4 a = ((const float4*)(Z + (size_t)s * 64))[l];
    float4 b = ((const float4*)(Z + (size_t)d * 64))[l];
    float p = a.x * b.x + a.y * b.y + a.z * b.z + a.w * b.w;
    #pragma unroll
    for (int off = 8; off > 0; off >>= 1) p += __shfl_xor(p, off, 16);
    if (l == 0) out[he] = p;
}

// ------------------------------------------------------------------- launcher
extern "C" void kernel_launch(void* const* d_in, const int* in_sizes, int n_in,
                              void* d_out, int out_size, void* d_ws, size_t ws_size,
                              hipStream_t stream) {
    const int IN_C = 128, HID = 128, OUT_C = 64;

    const float* x  = (const float*)d_in[0];
    const float* W1 = (const float*)d_in[1];
    const float* b1 = (const float*)d_in[2];
    const float* W2 = (const float*)d_in[3];
    const float* b2 = (const float*)d_in[4];
    const float* W3 = (const float*)d_in[5];
    const float* b3 = (const float*)d_in[6];
    const int* pos  = (const int*)d_in[7];
    const int* neg  = (const int*)d_in[8];

    const int N = in_sizes[0] / IN_C;
    const int E = in_sizes[7] / 2;

    const int* psrc = pos;     const int* pdst = pos + E;
    const int* nsrc = neg;     const int* ndst = neg + E;

    float* dinv = (float*)d_ws;                    // N floats
    float* A    = dinv + N;                        // N x 128 (GEMM output)
    float* B    = A + (size_t)N * HID;             // N x 128 (aggregated act.)
    float* out  = (float*)d_out;                   // [pos_score | neg_score]

    const int T = 256;
    // ---- gcn_norm: deg^(−1/2) with self-loops
    fill_ones_kernel<<<(N + T - 1) / T, T, 0, stream>>>(dinv, N);
    deg_count_kernel<<<(E + T - 1) / T, T, 0, stream>>>(pdst, E, dinv);
    rsqrt_kernel<<<(N + T - 1) / T, T, 0, stream>>>(dinv, N);

    const int Mt = (N + 15) / 16;
    auto gemm_blocks = [&](int cout) { return (Mt * (cout / 16) + 3) / 4; };
    const int t128_4 = N * (HID / 4);               // float4 count, C=128
    const int t64_4  = N * (OUT_C / 4);             // float4 count, C=64
    const int sc128  = (E + 7) / 8;                 // 8 edges / 256-thr block
    const int sc64   = (E + 15) / 16;               // 16 edges / 256-thr block
    const int dec    = (E + 15) / 16;               // 16 edges / 256-thr block

    // ---- layer 1: conv(x, W1) ; no input ReLU
    gemm_wmma_f32_kernel<false><<<gemm_blocks(HID), 128, 0, stream>>>(x, W1, A, N, IN_C, HID);
    self_bias_kernel<128><<<(t128_4 + T - 1) / T, T, 0, stream>>>(
        (const float4*)A, dinv, (const float4*)b1, (float4*)B, t128_4);
    scatter_edges_kernel<128><<<sc128, T, 0, stream>>>(A, dinv, psrc, pdst, B, E);

    // ---- layer 2: conv(relu(h1), W2) ; ReLU fused into GEMM input load
    gemm_wmma_f32_kernel<true><<<gemm_blocks(HID), 128, 0, stream>>>(B, W2, A, N, HID, HID);
    self_bias_kernel<128><<<(t128_4 + T - 1) / T, T, 0, stream>>>(
        (const float4*)A, dinv, (const float4*)b2, (float4*)B, t128_4);
    scatter_edges_kernel<128><<<sc128, T, 0, stream>>>(A, dinv, psrc, pdst, B, E);

    // ---- layer 3: conv(relu(h2), W3) -> z  (stored in B, N x 64)
    gemm_wmma_f32_kernel<true><<<gemm_blocks(OUT_C), 128, 0, stream>>>(B, W3, A, N, HID, OUT_C);
    self_bias_kernel<64><<<(t64_4 + T - 1) / T, T, 0, stream>>>(
        (const float4*)A, dinv, (const float4*)b3, (float4*)B, t64_4);
    scatter_edges_kernel<64><<<sc64, T, 0, stream>>>(A, dinv, psrc, pdst, B, E);

    // ---- decode: dot(z[src], z[dst]) for pos and neg edges
    decode_edges_kernel<<<dec, T, 0, stream>>>(B, psrc, pdst, out, E);
    decode_edges_kernel<<<dec, T, 0, stream>>>(B, nsrc, ndst, out + E, E);
}